// GetHeadsGemma2Attention_6760278524097
// MI455X (gfx1250) — compile-verified
//
#include <hip/hip_runtime.h>
#include <hip/hip_bf16.h>

// ---------------------------------------------------------------------------
// Gemma2 attention for MI455X (gfx1250), fp32 end-to-end via
// V_WMMA_F32_16X16X4_F32.  Memory-bound (268MB attn_weights output), so fp32
// matrix ops cost nothing vs bf16 and match reference numerics exactly.
// LDS tiles are filled with GLOBAL_LOAD_ASYNC_TO_LDS_B128 (ASYNCcnt path).
// Wave32 everywhere.
// ---------------------------------------------------------------------------

#define B_   2
#define S_   2048
#define D_   2304
#define H_   8
#define KV_  4
#define HD_  256
#define SCALE_   0.0625f        // 256^-0.5
#define MASKMIN_ (-1000000000.0f)

#define SCPAD 2052              // sc row stride (floats): 4 banks offset, 16B aligned
#define QSPAD 260               // qs row stride (floats): 4 banks offset, 16B aligned

typedef __attribute__((ext_vector_type(2))) float v2f;
typedef __attribute__((ext_vector_type(8))) float v8f;

static __device__ __forceinline__ v8f wmma_f32(v2f a, v2f b, v8f c) {
    // D = A(16x4) * B(4x16) + C(16x16), fp32.
    return __builtin_amdgcn_wmma_f32_16x16x4_f32(
        false, a, false, b, (short)0, c, false, false);
}

// Async global->LDS copy of 16 bytes per active lane (CDNA5 ASYNCcnt path).
// LDS offset = low 32 bits of the generic pointer (wave-relative per aperture).
static __device__ __forceinline__ void async_copy_b128(const float* gsrc,
                                                       float* ldst) {
    unsigned loff = (unsigned)(size_t)ldst;
    asm volatile("global_load_async_to_lds_b128 %0, %1, off"
                 :: "v"(loff), "v"(gsrc) : "memory");
}
static __device__ __forceinline__ void wait_async0() {
    asm volatile("s_wait_asynccnt 0x0" ::: "memory");
}

// ---------------------------------------------------------------------------
// Generic GEMM:  C = A(M x K) * W(N x K)^T,  M = B_*S_ rows.
// Block = 256 threads = 8 waves; block tile = 16 rows x 128 cols,
// each wave owns one 16x16 tile.  A and per-wave B tiles staged in LDS via
// async B128 copies.  NH/HDo are template params so the head-split epilogue
// index math is shifts/masks; NH==1 writes plain row-major (B,S,HDo).
// ---------------------------------------------------------------------------
template<int NH, int HDo>
__global__ __launch_bounds__(256)
void gemm_nt_kernel(const float* __restrict__ A, const float* __restrict__ W,
                    float* __restrict__ C, int Kdim)
{
    __shared__ float Ash[16][36];        // row stride 144B: 16B aligned, no bank conflict
    __shared__ float Bsh[8][16][36];

    const int tid  = threadIdx.x;
    const int w    = tid >> 5;
    const int lane = tid & 31;
    const int ml   = lane & 15;          // M (or N) index inside fragment
    const int sel  = (lane >> 4) << 1;   // lanes 0-15 -> K+{0,1}, 16-31 -> K+{2,3}

    const int row0 = blockIdx.y << 4;    // 16-row tile
    const int col0 = blockIdx.x << 7;    // 128-col block

    v8f acc = {};

    for (int kb = 0; kb < Kdim; kb += 32) {
        // stage A tile 16x32 floats = 128 x 16B chunks (async, coalesced)
        if (tid < 128) {
            int r = tid >> 3, c4 = (tid & 7) << 2;
            async_copy_b128(&A[(size_t)(row0 + r) * Kdim + kb + c4], &Ash[r][c4]);
        }
        // stage 8 wave-tiles of B (16x32 each) = 1024 x 16B chunks, 4/thread
        #pragma unroll
        for (int it = 0; it < 4; ++it) {
            int i = tid + (it << 8);
            int wv = i >> 7, rem = i & 127;
            int n = rem >> 3, c4 = (rem & 7) << 2;
            async_copy_b128(&W[(size_t)(col0 + (wv << 4) + n) * Kdim + kb + c4],
                            &Bsh[wv][n][c4]);
        }
        wait_async0();
        __syncthreads();

        #pragma unroll
        for (int j = 0; j < 8; ++j) {
            int k0 = (j << 2) + sel;
            v2f a, b;
            a.x = Ash[ml][k0];      a.y = Ash[ml][k0 + 1];
            b.x = Bsh[w][ml][k0];   b.y = Bsh[w][ml][k0 + 1];
            acc = wmma_f32(a, b, acc);
        }
        __syncthreads();
    }

    // epilogue: C/D layout -> VGPR r holds M = r (lanes 0-15) or 8+r (16-31)
    const int mhi = (lane >> 4) << 3;
    #pragma unroll
    for (int r = 0; r < 8; ++r) {
        int m = row0 + r + mhi;
        int b = m >> 11;                 // / S_
        int s = m & (S_ - 1);
        int n = col0 + (w << 4) + ml;
        if constexpr (NH == 1) {
            C[((size_t)b * S_ + s) * HDo + n] = acc[r];
        } else {
            int h = n / HDo;             // HDo = 256: shifts/masks
            int d = n % HDo;
            C[(((size_t)b * NH + h) * S_ + s) * HDo + d] = acc[r];
        }
    }
}

// ---------------------------------------------------------------------------
// RoPE, in-place on (B, NHt, S, HD).  One thread per (d, d+128) pair.
// ---------------------------------------------------------------------------
__global__ __launch_bounds__(256)
void rope_kernel(float* __restrict__ t, const float* __restrict__ cs,
                 const float* __restrict__ sn, int NHt, int total)
{
    int idx = blockIdx.x * 256 + threadIdx.x;
    if (idx >= total) return;
    int d = idx & 127;         idx >>= 7;
    int s = idx & (S_ - 1);    idx >>= 11;
    int h = idx % NHt;
    int b = idx / NHt;

    size_t cbase = ((size_t)b * S_ + s) * HD_;
    float c1 = cs[cbase + d],       s1 = sn[cbase + d];
    float c2 = cs[cbase + d + 128], s2 = sn[cbase + d + 128];

    size_t base = (((size_t)b * NHt + h) * S_ + s) * HD_;
    float x1 = t[base + d];
    float x2 = t[base + d + 128];
    t[base + d]       = x1 * c1 - x2 * s1;   // x*cos + rotate_half(x)*sin, d<128
    t[base + d + 128] = x2 * c2 + x1 * s2;   // d>=128 half
}

// ---------------------------------------------------------------------------
// Fused attention: one block per (b, h, 16-row q-tile).
// Phase B: scores (16 x 2048) via WMMA -> LDS (fully-masked tiles skipped,
//          next K tile prefetched).
// Phase C: in-place softmax (wave per 2 rows, shfl_xor reduction), stream
//          probabilities to attn_weights output.
// Phase D: attn = P*V via WMMA out of LDS, k-loop clipped at causal bound.
// LDS: 16*SCPAD*4 (~128KB) scores + 16*QSPAD*4 (~16KB) Q (WGP has 320KB).
// Row padding (+4 floats) -> A-fragment reads hit 16 distinct banks.
// ---------------------------------------------------------------------------
__global__ __launch_bounds__(256)
void attn_kernel(const float* __restrict__ Q, const float* __restrict__ K,
                 const float* __restrict__ V, float* __restrict__ wout,
                 float* __restrict__ mid)
{
    __shared__ float sc[16][SCPAD];
    __shared__ float qs[16][QSPAD];

    const int tid  = threadIdx.x;
    const int w    = tid >> 5;
    const int lane = tid & 31;
    const int ml   = lane & 15;
    const int sel  = (lane >> 4) << 1;
    const int mhi  = (lane >> 4) << 3;

    const int s0  = blockIdx.x << 4;
    const int h   = blockIdx.y;
    const int b   = blockIdx.z;
    const int kvh = h >> 1;                     // n_rep = H/KV = 2 (jnp.repeat)

    const float* Qg = Q + (((size_t)b * H_ + h) * S_ + s0) * HD_;
    const float* Kg = K + ((size_t)b * KV_ + kvh) * S_ * HD_;
    const float* Vg = V + ((size_t)b * KV_ + kvh) * S_ * HD_;

    // ---- load Q tile (16 x 256) via async B128: 1024 chunks, 4/thread ----
    #pragma unroll
    for (int it = 0; it < 4; ++it) {
        int i = tid + (it << 8);
        int r = i >> 6, c4 = (i & 63) << 2;
        async_copy_b128(&Qg[r * HD_ + c4], &qs[r][c4]);
    }
    wait_async0();
    __syncthreads();

    // ---- Phase B: scores; wave w owns key tiles [w*16, w*16+16) ----
    for (int t = w * 16; t < w * 16 + 16; ++t) {
        const int kt0 = t << 4;
        if (kt0 > s0 + 15) {                    // fully masked tile: no WMMA
            #pragma unroll
            for (int r = 0; r < 8; ++r)
                sc[r + mhi][kt0 + ml] = MASKMIN_;
            continue;
        }
        // prefetch next tile's K row for this lane (L2 warm-up)
        if (kt0 + 16 < S_)
            __builtin_prefetch(Kg + (size_t)(kt0 + 16 + ml) * HD_, 0, 0);

        v8f acc = {};
        const float* Kt = Kg + (size_t)(kt0 + ml) * HD_;
        #pragma unroll 4
        for (int j = 0; j < HD_ / 4; ++j) {
            int k0 = (j << 2) + sel;
            v2f a, bf;
            a.x  = qs[ml][k0];  a.y  = qs[ml][k0 + 1];
            bf.x = Kt[k0];      bf.y = Kt[k0 + 1];
            acc = wmma_f32(a, bf, acc);
        }
        #pragma unroll
        for (int r = 0; r < 8; ++r) {
            int qi = s0 + r + mhi;
            int ki = kt0 + ml;
            float v = acc[r] * SCALE_ + (ki <= qi ? 0.0f : MASKMIN_);
            sc[r + mhi][ki] = v;
        }
    }
    __syncthreads();

    // ---- Phase C: softmax rows; wave w handles rows 2w, 2w+1 ----
    for (int rr = w * 2; rr < w * 2 + 2; ++rr) {
        float mx = -3.4e38f;
        #pragma unroll 8
        for (int c = lane; c < S_; c += 32) mx = fmaxf(mx, sc[rr][c]);
        #pragma unroll
        for (int off = 16; off; off >>= 1) mx = fmaxf(mx, __shfl_xor(mx, off, 32));

        float sum = 0.0f;
        #pragma unroll 8
        for (int c = lane; c < S_; c += 32) {
            float e = __expf(sc[rr][c] - mx);   // masked -> exp(-1e9) == 0
            sc[rr][c] = e;
            sum += e;
        }
        #pragma unroll
        for (int off = 16; off; off >>= 1) sum += __shfl_xor(sum, off, 32);
        float inv = 1.0f / sum;

        float* Wr = wout + (((size_t)b * H_ + h) * S_ + (s0 + rr)) * S_;
        #pragma unroll 8
        for (int c = lane; c < S_; c += 32) {
            float p = sc[rr][c] * inv;
            sc[rr][c] = p;
            Wr[c] = p;
        }
    }
    __syncthreads();

    // ---- Phase D: attn = P * V; clip k at causal bound (P==0 beyond) ----
    const int kIter = (s0 + 16) >> 2;           // (s0+16)/4 WMMA k-steps
    for (int nt = w * 2; nt < w * 2 + 2; ++nt) {
        const int n0 = nt << 4;
        v8f acc = {};
        for (int j = 0; j < kIter; ++j) {
            int k0 = (j << 2) + sel;
            v2f a, bf;
            a.x  = sc[ml][k0];  a.y  = sc[ml][k0 + 1];
            bf.x = Vg[(size_t)k0 * HD_ + n0 + ml];
            bf.y = Vg[(size_t)(k0 + 1) * HD_ + n0 + ml];
            acc = wmma_f32(a, bf, acc);
        }
        #pragma unroll
        for (int r = 0; r < 8; ++r) {
            int s = s0 + r + mhi;
            mid[((size_t)b * S_ + s) * (H_ * HD_) + h * HD_ + n0 + ml] = acc[r];
        }
    }
}

// ---------------------------------------------------------------------------
extern "C" void kernel_launch(void* const* d_in, const int* in_sizes, int n_in,
                              void* d_out, int out_size, void* d_ws, size_t ws_size,
                              hipStream_t stream)
{
    const float* hidden = (const float*)d_in[0];
    const float* cosP   = (const float*)d_in[1];
    const float* sinP   = (const float*)d_in[2];
    // d_in[3] = attention_mask (causal; applied analytically in-kernel)
    const float* q_w    = (const float*)d_in[4];
    const float* k_w    = (const float*)d_in[5];
    const float* v_w    = (const float*)d_in[6];
    const float* o_w    = (const float*)d_in[7];

    float* ws   = (float*)d_ws;
    float* qBuf = ws;                                   // B*H*S*HD   = 8388608
    float* kBuf = qBuf + (size_t)B_ * H_  * S_ * HD_;   // B*KV*S*HD  = 4194304
    float* vBuf = kBuf + (size_t)B_ * KV_ * S_ * HD_;
    float* mid  = vBuf + (size_t)B_ * KV_ * S_ * HD_;   // B*S*(H*HD) = 8388608

    float* outAttn = (float*)d_out;                     // (B,S,D)
    float* outW    = outAttn + (size_t)B_ * S_ * D_;    // (B,H,S,S)

    dim3 blk(256);

    // QKV projections (head-split output layout)
    gemm_nt_kernel<H_, HD_><<<dim3((H_  * HD_) / 128, (B_ * S_) / 16), blk, 0, stream>>>(
        hidden, q_w, qBuf, D_);
    gemm_nt_kernel<KV_, HD_><<<dim3((KV_ * HD_) / 128, (B_ * S_) / 16), blk, 0, stream>>>(
        hidden, k_w, kBuf, D_);
    gemm_nt_kernel<KV_, HD_><<<dim3((KV_ * HD_) / 128, (B_ * S_) / 16), blk, 0, stream>>>(
        hidden, v_w, vBuf, D_);

    // RoPE on Q and K
    int totQ = B_ * H_  * S_ * (HD_ / 2);
    int totK = B_ * KV_ * S_ * (HD_ / 2);
    rope_kernel<<<(totQ + 255) / 256, blk, 0, stream>>>(qBuf, cosP, sinP, H_,  totQ);
    rope_kernel<<<(totK + 255) / 256, blk, 0, stream>>>(kBuf, cosP, sinP, KV_, totK);

    // Fused scores + softmax (+weights out) + P*V
    attn_kernel<<<dim3(S_ / 16, H_, B_), blk, 0, stream>>>(
        qBuf, kBuf, vBuf, outW, mid);

    // O projection (row-major (B,S,D) output)
    gemm_nt_kernel<1, D_><<<dim3(D_ / 128, (B_ * S_) / 16), blk, 0, stream>>>(
        mid, o_w, outAttn, H_ * HD_);
}